// HashEmbeddingBag_30597347016953
// MI455X (gfx1250) — compile-verified
//
#include <hip/hip_runtime.h>
#include <hip/hip_bf16.h>

typedef __attribute__((ext_vector_type(2))) float v2f;
typedef __attribute__((ext_vector_type(8))) float v8f;

// One wave (32 lanes) per output row b.
// Lane l owns output dims [4l, 4l+4) as a float4 -> 512B coalesced per embedding row.
__global__ __launch_bounds__(256) void hash_embedding_bag_kernel(
    const int* __restrict__ ids,      // (B, S) hashed ids (int32)
    const int* __restrict__ offsets,  // (B,) sorted window starts
    const float* __restrict__ emb,    // (n_emb, 128) fp32
    float* __restrict__ out,          // (B, 128) fp32
    int B, int S, int n_emb)
{
    const int lane = threadIdx.x & 31;
    const int wave = threadIdx.x >> 5;
    const int b = blockIdx.x * 8 + wave;
    if (b >= B) return;  // wave-uniform: EXEC stays all-ones inside live waves

    const int start = offsets[b];
    const int end   = (b + 1 < B) ? offsets[b + 1] : S;

    float ax = 0.f, ay = 0.f, az = 0.f, aw = 0.f;
    const long long idBase = (long long)b * S;

    // Process ids 32 at a time: lane-parallel fetch, shfl-broadcast, coalesced gather.
    for (int s0 = start; s0 < end; s0 += 32) {
        const int cnt = min(32, end - s0);
        int myid = (lane < cnt) ? ids[idBase + s0 + lane] : 0;
        // Prefetch this batch's embedding rows (global_prefetch_b8) to hide gather latency.
        {
            const unsigned hp = (unsigned)myid % (unsigned)n_emb;
            __builtin_prefetch(emb + (size_t)hp * 128 + 4 * lane, 0, 1);
        }
        for (int j = 0; j < cnt; ++j) {
            const int id = __shfl(myid, j, 32);
            const unsigned h = (unsigned)id % (unsigned)n_emb;
            const float4 e = ((const float4*)(emb + (size_t)h * 128))[lane];
            ax += e.x; ay += e.y; az += e.z; aw += e.w;
        }
    }

    // ---- 32-lane sum-of-squares reduction via V_WMMA_F32_16X16X4_F32 ----
    // p: per-lane partial. Fold halves: q_i lives at lanes i and 16+i (i=0..15).
    float p = ax * ax + ay * ay + az * az + aw * aw;
    float q = p + __shfl_xor(p, 16, 32);

    // Build A (16x4 fp32) so rows m=0..3 hold q_{4m+k}:
    //   lanes 0-15 supply (k=0,k=1), lanes 16-31 supply (k=2,k=3)  [ISA 16x4 A layout]
    const int m    = lane & 15;
    const int base = 4 * m + ((lane >= 16) ? 2 : 0);
    v2f a;
    a.x = __shfl(q, base & 31, 32);
    a.y = __shfl(q, (base + 1) & 31, 32);
    v2f ones;
    ones.x = 1.0f; ones.y = 1.0f;  // B = all-ones 4x16 -> D[m,n] = sum_k A[m,k]
    v8f c = {};
    c = __builtin_amdgcn_wmma_f32_16x16x4_f32(
        /*neg_a=*/false, a, /*neg_b=*/false, ones,
        /*c_mod=*/(short)0, c, /*reuse_a=*/false, /*reuse_b=*/false);
    // D layout: VGPR j, lanes 0-15 = row M=j. Rows 0..3 = quad sums of q.
    float total = c[0] + c[1] + c[2] + c[3];  // valid on lanes 0-15
    total = __shfl(total, 0, 32);             // broadcast lane 0 to whole wave

    // torch F.normalize: x / max(||x||, eps); empty window -> 0 (0 * 1/eps).
    const float scale = 1.0f / fmaxf(sqrtf(total), 1e-12f);
    float4 r;
    r.x = ax * scale; r.y = ay * scale; r.z = az * scale; r.w = aw * scale;
    ((float4*)(out + (size_t)b * 128))[lane] = r;
}

extern "C" void kernel_launch(void* const* d_in, const int* in_sizes, int n_in,
                              void* d_out, int out_size, void* d_ws, size_t ws_size,
                              hipStream_t stream) {
    const int*   ids     = (const int*)d_in[0];    // input_ids (B,S) int
    const int*   offsets = (const int*)d_in[1];    // offsets (B,) int
    const float* emb     = (const float*)d_in[2];  // embeddings (n_emb, D) fp32
    float*       out     = (float*)d_out;          // (B, D) fp32

    const int B     = in_sizes[1];
    const int S     = in_sizes[0] / B;
    const int D     = out_size / B;        // 128 by construction
    const int n_emb = in_sizes[2] / D;

    const int wavesPerBlock = 8;           // 256 threads = 8 wave32 per WGP launch
    const int blocks = (B + wavesPerBlock - 1) / wavesPerBlock;
    hash_embedding_bag_kernel<<<blocks, 256, 0, stream>>>(ids, offsets, emb, out, B, S, n_emb);
}